// RecommenderSystem_44203803410965
// MI455X (gfx1250) — compile-verified
//
#include <hip/hip_runtime.h>

#define D 128            // latent dim
#define FR_STRIDE 66     // fragment stride in dwords (64 payload + 2 pad -> conflict-free)

typedef __attribute__((ext_vector_type(2))) float v2f;
typedef __attribute__((ext_vector_type(8))) float v8f;

__device__ __forceinline__ float lrelu(float v) { return v > 0.0f ? v : 0.01f * v; }

// ---------------- elementwise helpers ----------------

__global__ void zero_f32(float* __restrict__ p, int n) {
  int i = blockIdx.x * blockDim.x + threadIdx.x;
  if (i < n) p[i] = 0.0f;
}

__global__ void deg_accum(const int* __restrict__ to_, float* __restrict__ deg, int E) {
  int e = blockIdx.x * blockDim.x + threadIdx.x;
  if (e < E) unsafeAtomicAdd(&deg[to_[e]], 1.0f);
}

__global__ void make_dinv(float* __restrict__ deg, int n) {
  int i = blockIdx.x * blockDim.x + threadIdx.x;
  if (i < n) {
    float d = deg[i];
    deg[i] = d > 0.0f ? rsqrtf(d) : 0.0f;
  }
}

// copy emb into d_out[0:N*128] and into out[:,0:128]
__global__ void copy_emb(const float* __restrict__ emb, float* __restrict__ out0,
                         float* __restrict__ out1, int total, int ostride) {
  int i = (blockIdx.x * blockDim.x + threadIdx.x) * 4;
  if (i < total) {
    float4 v = *(const float4*)(emb + i);
    *(float4*)(out0 + i) = v;
    int node = i >> 7, c = i & (D - 1);
    *(float4*)(out1 + (size_t)node * ostride + c) = v;
  }
}

// x_next = leaky_relu(agg) written into output slab (row stride = ostride)
__global__ void update_x(const float* __restrict__ agg, float* __restrict__ xout,
                         int total, int ostride) {
  int i = (blockIdx.x * blockDim.x + threadIdx.x) * 4;
  if (i < total) {
    float4 v = *(const float4*)(agg + i);
    v.x = lrelu(v.x); v.y = lrelu(v.y); v.z = lrelu(v.z); v.w = lrelu(v.w);
    int node = i >> 7, c = i & (D - 1);
    *(float4*)(xout + (size_t)node * ostride + c) = v;
  }
}

// stage W (128x128 row-major, k = row) into B-fragment layout:
// sWf[(kbq*8+nb)*FR_STRIDE + lane*2 + j] = W[kbq*4 + 2*(lane>=16) + j][nb*16 + (lane&15)]
__device__ __forceinline__ void stage_B_fragments(const float* __restrict__ W,
                                                  float* __restrict__ sWf,
                                                  int wave, int lane) {
  const int krow = (lane >> 4) << 1;   // 0 or 2
  const int ncol = lane & 15;
#pragma unroll 4
  for (int fr = wave; fr < 256; fr += 4) {
    int kbq = fr >> 3, nb = fr & 7;
    int k = (kbq << 2) + krow;
    int n = (nb << 4) + ncol;
    v2f v;
    v.x = W[k * D + n];
    v.y = W[(k + 1) * D + n];
    *(v2f*)(&sWf[fr * FR_STRIDE + lane * 2]) = v;
  }
}

// ---------------- WMMA node GEMM: lin1 = X@W + b ; agg = lin1 ----------------
// block = 128 threads (4 waves); each wave computes a 16x128 output tile.
__global__ void node_gemm_wmma(const float* __restrict__ X, int xstride,
                               const float* __restrict__ W, const float* __restrict__ bias,
                               float* __restrict__ lin1, float* __restrict__ agg, int nrows) {
  extern __shared__ float smem[];
  float* sWf = smem;                      // 256 * FR_STRIDE = 16896
  float* sAf = sWf + 256 * FR_STRIDE;     // 4 waves * 32 * FR_STRIDE = 8448
  float* sb  = sAf + 4 * 32 * FR_STRIDE;  // 128

  const int tid = threadIdx.x;
  const int wave = tid >> 5;
  const int lane = tid & 31;

  stage_B_fragments(W, sWf, wave, lane);
  if (tid < D) sb[tid] = bias[tid];

  const int rowBase = (blockIdx.x * 4 + wave) * 16;
  float* sAw = sAf + wave * 32 * FR_STRIDE;
  const bool active = rowBase < nrows;

  if (active) {
    // load 16 rows of X; lane covers cols 4*lane..4*lane+3 -> fragment kbq=lane,
    // slots r (lo pair) and r+16 (hi pair)
    for (int r = 0; r < 16; ++r) {
      int g = rowBase + r; if (g > nrows - 1) g = nrows - 1;
      float4 x = *(const float4*)(X + (size_t)g * xstride + lane * 4);
      v2f lo, hi;
      lo.x = x.x; lo.y = x.y;
      hi.x = x.z; hi.y = x.w;
      *(v2f*)(&sAw[lane * FR_STRIDE + r * 2])      = lo;
      *(v2f*)(&sAw[lane * FR_STRIDE + 32 + r * 2]) = hi;
    }
  }
  __syncthreads();

  if (active) {
    v8f acc[8] = {};
    for (int kbq = 0; kbq < 32; ++kbq) {
      v2f a = *(const v2f*)(&sAw[kbq * FR_STRIDE + lane * 2]);
#pragma unroll
      for (int nb = 0; nb < 8; ++nb) {
        v2f b = *(const v2f*)(&sWf[(kbq * 8 + nb) * FR_STRIDE + lane * 2]);
        acc[nb] = __builtin_amdgcn_wmma_f32_16x16x4_f32(
            false, a, false, b, (short)0, acc[nb], false, false);
      }
    }
    const int rlo = lane & 15;
#pragma unroll
    for (int nb = 0; nb < 8; ++nb) {
#pragma unroll
      for (int i = 0; i < 8; ++i) {
        int r = i + ((lane >> 4) << 3);
        int c = nb * 16 + rlo;
        int g = rowBase + r;
        if (g < nrows) {
          float val = acc[nb][i] + sb[c];
          size_t o = (size_t)g * D + c;
          lin1[o] = val;
          agg[o]  = val;     // seed agg with lin1
        }
      }
    }
  }
}

// ---------------- WMMA edge kernel ----------------
// msg = norm * (lin1[from] + (x[from]*x[to]) @ W2 + b2); atomic scatter into agg[to].
// block = 128 threads (4 waves); wave handles 16 edges x 128 cols.
__global__ void edge_msg_wmma(const int* __restrict__ from_, const int* __restrict__ to_,
                              const float* __restrict__ X, int xstride,
                              const float* __restrict__ lin1,
                              const float* __restrict__ W2, const float* __restrict__ b2,
                              const float* __restrict__ dinv,
                              float* __restrict__ agg, int E) {
  extern __shared__ float smem[];
  float* sWf = smem;                      // 16896
  float* sAf = sWf + 256 * FR_STRIDE;     // 8448
  float* sb  = sAf + 4 * 32 * FR_STRIDE;  // 128
  int*   sF  = (int*)(sb + D);            // 64
  int*   sT  = sF + 64;                   // 64
  float* sN  = (float*)(sT + 64);         // 64

  const int tid = threadIdx.x;
  const int wave = tid >> 5;
  const int lane = tid & 31;

  stage_B_fragments(W2, sWf, wave, lane);
  if (tid < D) sb[tid] = b2[tid];

  const int eBase = (blockIdx.x * 4 + wave) * 16;
  if (lane < 16) {
    int e = eBase + lane;
    int f = 0, t = 0; float nrm = 0.0f;
    if (e < E) { f = from_[e]; t = to_[e]; nrm = dinv[f] * dinv[t]; }
    sF[wave * 16 + lane] = f;
    sT[wave * 16 + lane] = t;
    sN[wave * 16 + lane] = nrm;
  }
  __syncthreads();

  // gather + elementwise product, written straight into A-fragment layout
  float* sAw = sAf + wave * 32 * FR_STRIDE;
  for (int r = 0; r < 16; ++r) {
    int f = sF[wave * 16 + r], t = sT[wave * 16 + r];
    float4 xf = *(const float4*)(X + (size_t)f * xstride + lane * 4);
    float4 xt = *(const float4*)(X + (size_t)t * xstride + lane * 4);
    v2f lo, hi;
    lo.x = xf.x * xt.x; lo.y = xf.y * xt.y;
    hi.x = xf.z * xt.z; hi.y = xf.w * xt.w;
    *(v2f*)(&sAw[lane * FR_STRIDE + r * 2])      = lo;
    *(v2f*)(&sAw[lane * FR_STRIDE + 32 + r * 2]) = hi;
  }
  __syncthreads();

  v8f acc[8] = {};
  for (int kbq = 0; kbq < 32; ++kbq) {
    v2f a = *(const v2f*)(&sAw[kbq * FR_STRIDE + lane * 2]);
#pragma unroll
    for (int nb = 0; nb < 8; ++nb) {
      v2f b = *(const v2f*)(&sWf[(kbq * 8 + nb) * FR_STRIDE + lane * 2]);
      acc[nb] = __builtin_amdgcn_wmma_f32_16x16x4_f32(
          false, a, false, b, (short)0, acc[nb], false, false);
    }
  }

  // epilogue: add lin1[from]+b2, scale by norm, atomic scatter into agg[to]
  const int rlo = lane & 15;
#pragma unroll
  for (int nb = 0; nb < 8; ++nb) {
#pragma unroll
    for (int i = 0; i < 8; ++i) {
      int r = i + ((lane >> 4) << 3);
      int c = nb * 16 + rlo;
      int f = sF[wave * 16 + r];
      int t = sT[wave * 16 + r];
      float nrm = sN[wave * 16 + r];       // 0 for out-of-range edges -> adds 0
      float lf = lin1[(size_t)f * D + c];
      float m = nrm * (acc[nb][i] + lf + sb[c]);
      unsafeAtomicAdd(&agg[(size_t)t * D + c], m);
    }
  }
}

// ---------------- host orchestration ----------------

extern "C" void kernel_launch(void* const* d_in, const int* in_sizes, int n_in,
                              void* d_out, int out_size, void* d_ws, size_t ws_size,
                              hipStream_t stream) {
  const int*   edges = (const int*)d_in[0];
  const float* emb   = (const float*)d_in[1];
  const float* W1    = (const float*)d_in[2];
  const float* b1    = (const float*)d_in[3];
  const float* W2    = (const float*)d_in[4];
  const float* b2    = (const float*)d_in[5];

  const int E = in_sizes[0] / 2;
  const int N = in_sizes[1] / D;
  const int L = in_sizes[2] / (D * D);

  const int* from_ = edges;
  const int* to_   = edges + E;

  float* out0 = (float*)d_out;               // emb copy [N,128]
  float* out1 = out0 + (size_t)N * D;        // concat [N, 128*(L+1)]
  const int ostride = D * (L + 1);           // 512

  float* ws   = (float*)d_ws;
  float* dinv = ws;
  float* lin1 = ws + (size_t)((N + 255) & ~255);
  float* agg  = lin1 + (size_t)N * D;

  // degree -> dinv
  zero_f32 <<<(N + 255) / 256, 256, 0, stream>>>(dinv, N);
  deg_accum<<<(E + 255) / 256, 256, 0, stream>>>(to_, dinv, E);
  make_dinv<<<(N + 255) / 256, 256, 0, stream>>>(dinv, N);

  const int total = N * D;
  copy_emb<<<(total / 4 + 255) / 256, 256, 0, stream>>>(emb, out0, out1, total, ostride);

  const int nodeBlocks = ((N + 15) / 16 + 3) / 4;
  const int edgeBlocks = (E + 63) / 64;
  const size_t smemNode = (size_t)(256 * FR_STRIDE + 4 * 32 * FR_STRIDE + D) * sizeof(float);
  const size_t smemEdge = (size_t)(256 * FR_STRIDE + 4 * 32 * FR_STRIDE + D + 192) * sizeof(float);

  const float* xp = emb;
  int xs = D;
  for (int l = 0; l < L; ++l) {
    node_gemm_wmma<<<nodeBlocks, 128, smemNode, stream>>>(
        xp, xs, W1 + (size_t)l * D * D, b1 + (size_t)l * D, lin1, agg, N);
    edge_msg_wmma<<<edgeBlocks, 128, smemEdge, stream>>>(
        from_, to_, xp, xs, lin1, W2 + (size_t)l * D * D, b2 + (size_t)l * D,
        dinv, agg, E);
    float* xnext = out1 + (size_t)(l + 1) * D;
    update_x<<<(total / 4 + 255) / 256, 256, 0, stream>>>(agg, xnext, total, ostride);
    xp = xnext;
    xs = ostride;
  }
}